// MultiLoss_87668872446687
// MI455X (gfx1250) — compile-verified
//
#include <hip/hip_runtime.h>
#include <hip/hip_bf16.h>

typedef __attribute__((ext_vector_type(2))) float v2f;
typedef __attribute__((ext_vector_type(8))) float v8f;

#define ALPHA_C 0.25f
#define IOU_T   0.5f
#define WX 10.0f
#define WY 10.0f
#define WW 5.0f
#define WH 5.0f
#define WT 15.0f

// ---------------- device helpers ----------------

__device__ __forceinline__ float smoothL1(float d) {
    d = fabsf(d);
    const float beta = 1.0f / 9.0f;
    return (d < beta) ? (0.5f * d * d / beta) : (d - 0.5f * beta);
}

__device__ __forceinline__ float balancedL1(float d) {
    d = fabsf(d);
    const float beta = 0.5f, al = 0.5f, ga = 0.9f;
    const float b = 5.0496474644129465f;          // e^(ga/al) - 1
    return (d < beta)
        ? (al / b * (b * d + 1.0f) * logf(b * d / beta + 1.0f) - al * d)
        : (ga * d + ga / b - al * beta);
}

__device__ __forceinline__ void boxCorners(const float* bx, float* px, float* py) {
    float c = cosf(bx[4]), s = sinf(bx[4]);
    const float ddx[4] = {-0.5f, 0.5f, 0.5f, -0.5f};
    const float ddy[4] = {-0.5f, -0.5f, 0.5f, 0.5f};
#pragma unroll
    for (int i = 0; i < 4; i++) {
        float dx = ddx[i] * bx[2], dy = ddy[i] * bx[3];
        px[i] = bx[0] + dx * c - dy * s;
        py[i] = bx[1] + dx * s + dy * c;
    }
}

// exact rotated-rectangle IoU via Sutherland-Hodgman clipping (matches reference)
__device__ float rboxIoU(const float* b1, const float* b2) {
    float p2x[4], p2y[4];
    float px[16], py[16], qx[16], qy[16];
    boxCorners(b1, px, py);
    boxCorners(b2, p2x, p2y);
    int n = 4;
    for (int e = 0; e < 4 && n > 0; e++) {
        float ax = p2x[e], ay = p2y[e];
        float dx = p2x[(e + 1) & 3] - ax, dyv = p2y[(e + 1) & 3] - ay;
        int m = 0;
        for (int i = 0; i < n; i++) {
            float cx = px[i], cy = py[i];
            int i1 = (i + 1 == n) ? 0 : i + 1;
            float nx = px[i1], ny = py[i1];
            float sc = dx * (cy - ay) - dyv * (cx - ax);
            float sn = dx * (ny - ay) - dyv * (nx - ax);
            bool kin = (sc >= 0.0f), nin = (sn >= 0.0f);
            if (kin && m < 16) { qx[m] = cx; qy[m] = cy; m++; }
            if ((kin != nin) && m < 16) {
                float den = sc - sn;
                float t = sc / ((den == 0.0f) ? 1.0f : den);
                qx[m] = cx + t * (nx - cx);
                qy[m] = cy + t * (ny - cy);
                m++;
            }
        }
        for (int i = 0; i < m; i++) { px[i] = qx[i]; py[i] = qy[i]; }
        n = m;
    }
    float cr = 0.0f;
    for (int i = 0; i < n; i++) {
        int i1 = (i + 1 == n) ? 0 : i + 1;
        cr += px[i] * py[i1] - px[i1] * py[i];
    }
    float inter = 0.5f * fabsf(cr);
    float uni = b1[2] * b1[3] + b2[2] * b2[3] - inter;
    return inter / fmaxf(uni, 1e-9f);
}

// axis-aligned IoU of min-area squares (the cheap gate)
__device__ __forceinline__ float sqIoU(const float* a5, const float* g5) {
    float sa = fmaxf(a5[2], a5[3]) * 0.5f, sg = fmaxf(g5[2], g5[3]) * 0.5f;
    float w = fminf(a5[0] + sa, g5[0] + sg) - fmaxf(a5[0] - sa, g5[0] - sg);
    float h = fminf(a5[1] + sa, g5[1] + sg) - fmaxf(a5[1] - sa, g5[1] - sg);
    w = fmaxf(w, 0.0f); h = fmaxf(h, 0.0f);
    float inter = w * h;
    float A1 = 4.0f * sa * sa, A2 = 4.0f * sg * sg;
    return inter / fmaxf(A1 + A2 - inter, 1e-9f);
}

// ---------------- kernels ----------------

__global__ void k_init(int* forced, unsigned long long* gmax, int nF, int nG) {
    int t = blockIdx.x * blockDim.x + threadIdx.x;
    if (t < nF) forced[t] = 0;
    if (t < nG) gmax[t] = 0ull;
}

// one thread per (b, anchor): IoU row, row max/argmax, wave-reduced column max
__global__ void k_pass1(const float* __restrict__ anchors, const float* __restrict__ ann,
                        float* __restrict__ iouMax, int* __restrict__ iouArg,
                        unsigned long long* __restrict__ gmax, int A, int G) {
    __shared__ float sann[24 * 6];
    int b = blockIdx.y;
    int a = blockIdx.x * blockDim.x + threadIdx.x;
    if (threadIdx.x < G * 6) sann[threadIdx.x] = ann[b * G * 6 + threadIdx.x];
    __syncthreads();
    __builtin_prefetch(&anchors[(b * A + a) * 5], 0, 3);   // global_prefetch_b8
    float an[5];
#pragma unroll
    for (int k = 0; k < 5; k++) an[k] = anchors[(b * A + a) * 5 + k];
    float best = -1e30f;
    int barg = 0;
    for (int g = 0; g < G; g++) {
        const float* gb = &sann[g * 6];
        bool valid = (gb[5] != -1.0f);        // uniform across wave
        float v;
        if (valid) {
            float ind = sqIoU(an, gb);
            v = (ind >= 0.1f) ? rboxIoU(an, gb) : 0.0f;
        } else {
            v = -1.0f;
        }
        if (v > best) { best = v; barg = g; }  // strict > keeps first max (jnp argmax)
        if (valid) {
            // pack (iou, ~anchor) so atomicMax reproduces smallest-index tie-break
            unsigned long long key =
                (((unsigned long long)__float_as_uint(v)) << 32) |
                (unsigned long long)(0xFFFFFFFFu - (unsigned)a);
#pragma unroll
            for (int off = 16; off > 0; off >>= 1) {
                unsigned long long o = __shfl_xor(key, off, 32);
                if (o > key) key = o;
            }
            if ((threadIdx.x & 31) == 0) atomicMax(&gmax[b * G + g], key);
        }
    }
    iouMax[b * A + a] = best;
    iouArg[b * A + a] = barg;
}

// force-match: gt whose best anchor IoU < 0.5 forces that anchor positive
__global__ void k_pass2(const float* __restrict__ ann,
                        const unsigned long long* __restrict__ gmax,
                        int* __restrict__ forced, int A, int G, int B) {
    int t = blockIdx.x * blockDim.x + threadIdx.x;
    if (t >= B * G) return;
    int b = t / G, g = t % G;
    if (ann[(b * G + g) * 6 + 5] == -1.0f) return;
    unsigned long long key = gmax[b * G + g];
    float mg = __uint_as_float((unsigned)(key >> 32));
    if (mg < IOU_T) {
        unsigned aidx = 0xFFFFFFFFu - (unsigned)(key & 0xFFFFFFFFull);
        if (aidx < (unsigned)A) forced[b * A + aidx] = 1;
    }
}

// per-anchor losses + deterministic block partial sums {posCount, cls, r1, r2}
__global__ void k_pass3(const float* __restrict__ cls, const float* __restrict__ reg,
                        const float* __restrict__ lmk, const float* __restrict__ anchors,
                        const float* __restrict__ ann, const float* __restrict__ ls,
                        const float* __restrict__ iouMax, const int* __restrict__ iouArg,
                        const int* __restrict__ forced, float* __restrict__ partials,
                        int A, int G, int C) {
    __shared__ float sann[24 * 6];
    __shared__ float sls[24 * 4];
    __shared__ float red[32];
    int b = blockIdx.y;
    int a = blockIdx.x * blockDim.x + threadIdx.x;
    if (threadIdx.x < G * 6) sann[threadIdx.x] = ann[b * G * 6 + threadIdx.x];
    if (threadIdx.x < G * 4) sls[threadIdx.x] = ls[b * G * 4 + threadIdx.x];
    __syncthreads();
    int idx = b * A + a;
    float im = iouMax[idx];
    int ag = iouArg[idx];
    bool pos = (im >= IOU_T) || (forced[idx] != 0);
    const float* gb = &sann[ag * 6];
    int lab = (int)gb[5];

    float clsSum = 0.0f;
#pragma unroll
    for (int c = 0; c < 8; c++) {
        float p = cls[idx * C + c];
        p = fminf(fmaxf(p, 1e-4f), 1.0f - 1e-4f);
        float tgt = pos ? ((c == lab) ? 1.0f : 0.0f)
                        : ((im < IOU_T - 0.1f) ? 0.0f : -1.0f);
        if (tgt != -1.0f) {
            bool one = (tgt == 1.0f);
            float af = one ? ALPHA_C : (1.0f - ALPHA_C);
            float pt = one ? (1.0f - p) : p;
            float fw = af * pt * pt;
            float bce = -(tgt * logf(p + 1e-6f) + (1.0f - tgt) * logf(1.0f - p + 1e-6f));
            clsSum += fw * bce;
        }
    }

    float r1 = 0.0f, r2 = 0.0f;
    if (pos) {
        float an[5], rg[5];
#pragma unroll
        for (int k = 0; k < 5; k++) { an[k] = anchors[idx * 5 + k]; rg[k] = reg[idx * 5 + k]; }
        float ew = fmaxf(an[2], 1.0f), eh = fmaxf(an[3], 1.0f);
        float gw = fmaxf(gb[2], 1.0f), gh = fmaxf(gb[3], 1.0f);
        float t1x = WX * (gb[0] - an[0]) / ew;
        float t1y = WY * (gb[1] - an[1]) / eh;
        float t1w = WW * logf(gw / ew);
        float t1h = WH * logf(gh / eh);
        float ta = tanf(an[4]);
        float t1t = WT * (tanf(gb[4]) - ta);
        float tg = rg[4] / WT + ta;                       // lmr5pangle
        if (fabsf(tg) < 1e-4f) tg = (tg < 0.0f) ? -1e-4f : 1e-4f;
        float t22 = WT * (-1.0f / tg - ta);
        float l1 = smoothL1(t1x - rg[0]), l2 = smoothL1(t1y - rg[1]);
        float l3 = smoothL1(t1w - rg[2]), l4 = smoothL1(t1h - rg[3]);
        float l5 = smoothL1(t1w - rg[3]), l6 = smoothL1(t1h - rg[2]);
        float l7 = smoothL1(t1t - rg[4]), l8 = smoothL1(t1t - t22);
        r1 = fminf(l1 + l2 + l3 + l4 + l7, l1 + l2 + l5 + l6 + l8);

        const float* lp = &sls[ag * 4];
        float t2a = WX * (lp[0] - an[0]) / ew;
        float t2b = WY * (lp[1] - an[1]) / eh;
        float t2c = WX * (lp[2] - an[0]) / ew;
        float t2d = WY * (lp[3] - an[1]) / eh;
        r2 = balancedL1(lmk[idx * 4 + 0] - t2a) + balancedL1(lmk[idx * 4 + 1] - t2b)
           + balancedL1(lmk[idx * 4 + 2] - t2c) + balancedL1(lmk[idx * 4 + 3] - t2d);
    }

    float vals[4] = { pos ? 1.0f : 0.0f, clsSum, r1, r2 };
    int lane = threadIdx.x & 31;
    int wid = threadIdx.x >> 5;
#pragma unroll
    for (int q = 0; q < 4; q++) {
        float v = vals[q];
#pragma unroll
        for (int off = 16; off > 0; off >>= 1) v += __shfl_xor(v, off, 32);
        if (lane == 0) red[q * 8 + wid] = v;
    }
    __syncthreads();
    if (threadIdx.x == 0) {
#pragma unroll
        for (int q = 0; q < 4; q++) {
            float s = 0.0f;
            for (int w = 0; w < 8; w++) s += red[q * 8 + w];
            partials[(b * gridDim.x + blockIdx.x) * 4 + q] = s;
        }
    }
}

// final: 64-way partial sums on the matrix pipe (A=data 16x4, B=ones -> row sums),
// then denominators + batch mean. Launched with exactly 32 threads (one wave,
// EXEC all-ones around the WMMA as the ISA requires).
__global__ void k_pass4(const float* __restrict__ partials, const float* __restrict__ ann,
                        float* __restrict__ out, int nb, int G, int B) {
    int lane = threadIdx.x;
    v2f bones; bones[0] = 1.0f; bones[1] = 1.0f;   // all-ones B is layout-invariant
    int m = lane & 15;
    int kbase = (lane < 16) ? 0 : 2;               // 16x4 f32 A-matrix VGPR layout
    float tot[2][4];
    for (int b = 0; b < 2; b++) {
        for (int q = 0; q < 4; q++) {
            v2f av;
            av[0] = partials[(b * nb + m * 4 + kbase + 0) * 4 + q];
            av[1] = partials[(b * nb + m * 4 + kbase + 1) * 4 + q];
            v8f c = {};
            c = __builtin_amdgcn_wmma_f32_16x16x4_f32(
                false, av, false, bones, (short)0, c, false, false);
            float s = 0.0f;
#pragma unroll
            for (int r = 0; r < 8; r++) s += c[r];  // rows 0-7 (lane0) / 8-15 (lane16)
            tot[b][q] = __shfl(s, 0, 32) + __shfl(s, 16, 32);
        }
    }
    if (lane == 0) {
        float clsm = 0.0f, r1m = 0.0f, r2m = 0.0f;
        for (int b = 0; b < B; b++) {
            bool hasGt = false;
            for (int g = 0; g < G; g++)
                if (ann[(b * G + g) * 6 + 5] != -1.0f) hasGt = true;
            float cnt = tot[b][0];
            float denom = fmaxf(cnt, 1.0f);
            clsm += hasGt ? tot[b][1] / denom : 0.0f;
            r1m += (hasGt && cnt > 0.0f) ? tot[b][2] / (5.0f * denom) : 0.0f;
            r2m += (hasGt && cnt > 0.0f) ? tot[b][3] / (4.0f * denom) : 0.0f;
        }
        out[0] = clsm / (float)B;
        out[1] = r1m / (float)B;
        out[2] = r2m / (float)B;
    }
}

// ---------------- launch ----------------

extern "C" void kernel_launch(void* const* d_in, const int* in_sizes, int n_in,
                              void* d_out, int out_size, void* d_ws, size_t ws_size,
                              hipStream_t stream) {
    const float* cls = (const float*)d_in[0];
    const float* reg = (const float*)d_in[1];
    const float* anc = (const float*)d_in[2];
    const float* ann = (const float*)d_in[3];
    const float* lmk = (const float*)d_in[4];
    const float* ls  = (const float*)d_in[5];
    const int B = 2, G = 24, C = 8;
    const int A = in_sizes[1] / (B * 5);           // 16384
    float* outp = (float*)d_out;

    char* ws = (char*)d_ws;
    float* iouMax = (float*)ws;                    ws += (size_t)B * A * sizeof(float);
    int*   iouArg = (int*)ws;                      ws += (size_t)B * A * sizeof(int);
    int*   forced = (int*)ws;                      ws += (size_t)B * A * sizeof(int);
    unsigned long long* gmax = (unsigned long long*)ws; ws += (size_t)B * G * sizeof(unsigned long long);
    float* partials = (float*)ws;

    const int TPB = 256;
    const int nb = A / TPB;                        // blocks per image (64)

    k_init<<<dim3((B * A + TPB - 1) / TPB), dim3(TPB), 0, stream>>>(forced, gmax, B * A, B * G);
    k_pass1<<<dim3(nb, B), dim3(TPB), 0, stream>>>(anc, ann, iouMax, iouArg, gmax, A, G);
    k_pass2<<<dim3(1), dim3(64), 0, stream>>>(ann, gmax, forced, A, G, B);
    k_pass3<<<dim3(nb, B), dim3(TPB), 0, stream>>>(cls, reg, lmk, anc, ann, ls,
                                                   iouMax, iouArg, forced, partials, A, G, C);
    k_pass4<<<dim3(1), dim3(32), 0, stream>>>(partials, ann, outp, nb, G, B);
}